// Sampler_50921132262004
// MI455X (gfx1250) — compile-verified
//
#include <hip/hip_runtime.h>
#include <hip/hip_bf16.h>
#include <math.h>

// ---------------------------------------------------------------------------
// Sampler (temperature -> softmax -> sort -> top-k(<=64)/top-p/min-p -> draw)
// B=128 rows, V=128000 vocab. Memory-bound: one async-staged streaming pass
// (K1) + tiny per-row merge/sample (K2).
// ---------------------------------------------------------------------------

#define TPB     256
#define BATCH   128
#define VOCAB   128000
#define SPLITS  32
#define CHUNK   (VOCAB / SPLITS)   // 4000 floats = 16000 B per block
#define NB128   (CHUNK / 4)        // 1000 x b128 async transfers per block
#define KTOP    64
#define BINS    4096
#define SEGBINS (BINS / TPB)       // 16 bins per thread segment
#define LCAP    512                // per-chunk candidate cap

// Monotone key: order of keys == order of floats (desc float -> desc key)
__device__ __forceinline__ unsigned fkey(float f) {
  unsigned u = __float_as_uint(f);
  return (u & 0x80000000u) ? ~u : (u | 0x80000000u);
}

// LDS byte address of a shared-memory pointer (flat low 32 bits == DS address)
__device__ __forceinline__ unsigned lds_addr(const void* p) {
  return (unsigned)(unsigned long long)p;
}

// CDNA5 async global->LDS copy (ASYNCcnt tracked), GV addressing mode.
__device__ __forceinline__ void async_load_b128(unsigned lds_byte, const void* gaddr) {
  asm volatile("global_load_async_to_lds_b128 %0, %1, off"
               :: "v"(lds_byte), "v"(gaddr)
               : "memory");
}

__device__ __forceinline__ void wait_async0() {
  asm volatile("s_wait_asynccnt 0" ::: "memory");
}

// ---------------------------------------------------------------------------
// K1: per (row, chunk): async-stage chunk to LDS; partial max + exp-sum;
// 4096-bin histogram radix-select of chunk top-64 (sorted, stable by index).
// ---------------------------------------------------------------------------
__global__ void __launch_bounds__(TPB) k_chunk_select(
    const float* __restrict__ logits,
    const float* __restrict__ temps,
    float* __restrict__ cand_val, int* __restrict__ cand_idx,
    float* __restrict__ pmax, float* __restrict__ psum)
{
  __shared__ __align__(16) float sdata[CHUNK];
  __shared__ unsigned shist[BINS];
  __shared__ unsigned ssuf[TPB];
  __shared__ float    sred[TPB];
  __shared__ float    scv[LCAP];
  __shared__ int      sci[LCAP];
  __shared__ unsigned scnt;
  __shared__ unsigned sthresh;

  const int bid = blockIdx.x;
  const int tid = threadIdx.x;
  const int row = bid / SPLITS;
  const int chk = bid % SPLITS;
  const float* gp = logits + (size_t)row * VOCAB + (size_t)chk * CHUNK;

  for (int i = tid; i < BINS; i += TPB) shist[i] = 0u;
  if (tid == 0) scnt = 0u;

  // ---- async stage: 16 KB chunk -> LDS (b128 per lane) ----
  const unsigned sbase = lds_addr(&sdata[0]);
  for (int j = tid; j < NB128; j += TPB)
    async_load_b128(sbase + (unsigned)j * 16u, (const void*)(gp + j * 4));
  wait_async0();
  __syncthreads();

  const float invT = 1.0f / temps[row];

  // ---- pass 1: chunk max + histogram ----
  float m = -INFINITY;
  for (int i = tid; i < CHUNK; i += TPB) {
    float v = sdata[i];
    m = fmaxf(m, v);
    atomicAdd(&shist[fkey(v) >> 20], 1u);
  }
  sred[tid] = m;
  __syncthreads();
  for (int off = TPB / 2; off > 0; off >>= 1) {
    if (tid < off) sred[tid] = fmaxf(sred[tid], sred[tid + off]);
    __syncthreads();
  }
  m = sred[0];
  __syncthreads();

  // ---- pass 2: exp-sum relative to chunk max ----
  float s = 0.f;
  for (int i = tid; i < CHUNK; i += TPB)
    s += __expf((sdata[i] - m) * invT);
  sred[tid] = s;
  __syncthreads();
  for (int off = TPB / 2; off > 0; off >>= 1) {
    if (tid < off) sred[tid] += sred[tid + off];
    __syncthreads();
  }
  if (tid == 0) { pmax[bid] = m; psum[bid] = sred[0]; }

  // ---- suffix counts over 16-bin segments -> 64th-value threshold bin ----
  unsigned seg = 0;
  for (int b = 0; b < SEGBINS; ++b) seg += shist[tid * SEGBINS + b];
  ssuf[tid] = seg;
  __syncthreads();
  for (int off = 1; off < TPB; off <<= 1) {
    unsigned add = (tid + off < TPB) ? ssuf[tid + off] : 0u;
    __syncthreads();
    ssuf[tid] += add;
    __syncthreads();
  }
  unsigned above = (tid + 1 < TPB) ? ssuf[tid + 1] : 0u;
  if (ssuf[tid] >= KTOP && above < KTOP) {      // unique crossing segment
    unsigned run = above;
    int b = tid * SEGBINS + SEGBINS - 1;
    for (;; --b) {
      run += shist[b];
      if (run >= KTOP || b == tid * SEGBINS) break;
    }
    sthresh = (unsigned)b << 20;                // lower key bound of bin b
  }
  __syncthreads();
  const unsigned th = sthresh;

  // ---- collect candidates >= threshold bin ----
  for (int i = tid; i < CHUNK; i += TPB) {
    float v = sdata[i];
    if (fkey(v) >= th) {
      unsigned p = atomicAdd(&scnt, 1u);
      if (p < LCAP) { scv[p] = v; sci[p] = chk * CHUNK + i; }
    }
  }
  __syncthreads();
  const int n = (scnt < (unsigned)LCAP) ? (int)scnt : LCAP;

  // default-fill the 64 output slots (also clears poisoned workspace)
  for (int t = tid; t < KTOP; t += TPB) {
    cand_val[bid * KTOP + t] = -INFINITY;
    cand_idx[bid * KTOP + t] = 0x7FFFFFFF;
  }
  __syncthreads();

  // ---- rank-count exact sorted top-64 (value desc, index asc) ----
  for (int i = tid; i < n; i += TPB) {
    float vi = scv[i]; int ii = sci[i];
    int rank = 0;
    for (int j = 0; j < n; ++j) {
      float vj = scv[j];
      rank += (vj > vi) || (vj == vi && sci[j] < ii);
    }
    if (rank < KTOP) {
      cand_val[bid * KTOP + rank] = vi;
      cand_idx[bid * KTOP + rank] = ii;
    }
  }
}

// ---------------------------------------------------------------------------
// K2: per row: merge 32 sorted top-64 lists -> global top-64; merge (m,s)
// partials -> (M, Z); apply top-k/top-p/min-p masks; inverse-CDF sample.
// ---------------------------------------------------------------------------
__global__ void __launch_bounds__(TPB) k_merge_sample(
    const float* __restrict__ cand_val, const int* __restrict__ cand_idx,
    const float* __restrict__ pmax, const float* __restrict__ psum,
    const float* __restrict__ temps, const int* __restrict__ top_ks,
    const float* __restrict__ top_ps, const float* __restrict__ min_ps,
    const float* __restrict__ uniform, int* __restrict__ out)
{
  const int NC = SPLITS * KTOP;   // 2048 candidates per row
  __shared__ float vals[NC]; __shared__ int idxs[NC];
  __shared__ float cv[NC];   __shared__ int ci[NC];
  __shared__ float top_v[KTOP]; __shared__ int top_i[KTOP];
  __shared__ unsigned scnt;
  __shared__ float sM, sZ, sPrune;

  const int row = blockIdx.x;
  const int tid = threadIdx.x;
  const float invT = 1.0f / temps[row];
  const int cbase = row * NC;

  for (int j = tid; j < NC; j += TPB) {
    vals[j] = cand_val[cbase + j];
    idxs[j] = cand_idx[cbase + j];
  }
  if (tid == 0) scnt = 0u;
  __syncthreads();

  if (tid == 0) {
    // prune threshold: max over chunks of their 64th value -> global 64th >= tp,
    // so every global top-64 element survives the >= tp filter.
    float tp = -INFINITY;
    for (int s = 0; s < SPLITS; ++s) tp = fmaxf(tp, vals[s * KTOP + KTOP - 1]);
    sPrune = tp;
    // softmax denominator: online-softmax merge of chunk partials
    float M = -INFINITY;
    for (int s = 0; s < SPLITS; ++s) M = fmaxf(M, pmax[row * SPLITS + s]);
    float Z = 0.f;
    for (int s = 0; s < SPLITS; ++s)
      Z += psum[row * SPLITS + s] * __expf((pmax[row * SPLITS + s] - M) * invT);
    sM = M; sZ = Z;
  }
  __syncthreads();
  const float prune = sPrune;

  for (int j = tid; j < NC; j += TPB) {
    float v = vals[j];
    if (v >= prune) {
      unsigned p = atomicAdd(&scnt, 1u);
      cv[p] = v; ci[p] = idxs[j];
    }
  }
  for (int t = tid; t < KTOP; t += TPB) { top_v[t] = -INFINITY; top_i[t] = 0; }
  __syncthreads();
  const int n = (int)scnt;

  // rank-count global sorted top-64 among ~100-150 survivors
  for (int i = tid; i < n; i += TPB) {
    float vi = cv[i]; int ii = ci[i];
    int rank = 0;
    for (int j = 0; j < n; ++j) {
      float vj = cv[j];
      rank += (vj > vi) || (vj == vi && ci[j] < ii);
    }
    if (rank < KTOP) { top_v[rank] = vi; top_i[rank] = ii; }
  }
  __syncthreads();

  if (tid == 0) {
    const float M = sM, Z = sZ;
    const float topp = top_ps[row];
    const int   topk = top_ks[row];
    const float minp = min_ps[row];
    const float u    = uniform[row];

    float p[KTOP];
    float cum = 0.f;
    for (int j = 0; j < KTOP; ++j) {
      float pj = __expf((top_v[j] - M) * invT) / Z;   // -inf pad -> 0
      float cbefore = cum;                            // cumsum - p
      cum += pj;
      bool keep = (cbefore <= topp) && (j < topk);
      p[j] = keep ? pj : 0.f;
    }
    const float minth = p[0] * minp;                  // p[0]=1/Z always kept
    float mx = 0.f;
    for (int j = 0; j < KTOP; ++j) {
      if (p[j] < minth) p[j] = 0.f;                   // min-p after k/p masks
      mx = fmaxf(mx, p[j]);
    }
    const float inv_mx = 1.0f / mx;                   // torch divides by row max
    float total = 0.f;
    for (int j = 0; j < KTOP; ++j) total += p[j] * inv_mx;
    const float uu = u * total;
    float cdf = 0.f;
    int sampled = 0;
    for (int j = 0; j < KTOP; ++j) { cdf += p[j] * inv_mx; sampled += (cdf < uu); }
    if (sampled > KTOP - 1) sampled = KTOP - 1;
    out[row] = top_i[sampled];
  }
}

// ---------------------------------------------------------------------------
extern "C" void kernel_launch(void* const* d_in, const int* in_sizes, int n_in,
                              void* d_out, int out_size, void* d_ws, size_t ws_size,
                              hipStream_t stream) {
  (void)in_sizes; (void)n_in; (void)out_size; (void)ws_size;
  const float* logits = (const float*)d_in[0];
  const float* temps  = (const float*)d_in[1];
  const int*   topks  = (const int*)d_in[2];
  const float* topps  = (const float*)d_in[3];
  const float* minps  = (const float*)d_in[4];
  const float* unif   = (const float*)d_in[5];

  char* ws = (char*)d_ws;
  float* cand_val = (float*)ws; ws += (size_t)BATCH * SPLITS * KTOP * sizeof(float);
  int*   cand_idx = (int*)ws;   ws += (size_t)BATCH * SPLITS * KTOP * sizeof(int);
  float* pmax     = (float*)ws; ws += (size_t)BATCH * SPLITS * sizeof(float);
  float* psum     = (float*)ws; ws += (size_t)BATCH * SPLITS * sizeof(float);

  k_chunk_select<<<dim3(BATCH * SPLITS), dim3(TPB), 0, stream>>>(
      logits, temps, cand_val, cand_idx, pmax, psum);
  k_merge_sample<<<dim3(BATCH), dim3(TPB), 0, stream>>>(
      cand_val, cand_idx, pmax, psum, temps, topks, topps, minps, unif,
      (int*)d_out);
}